// HanningLoss_5652176962068
// MI455X (gfx1250) — compile-verified
//
#include <hip/hip_runtime.h>
#include <hip/hip_bf16.h>

#define BB 48
#define HH 512
#define WW 512
#define KK 33

typedef __attribute__((ext_vector_type(2))) float v2f;
typedef __attribute__((ext_vector_type(8))) float v8f;

// Stable softplus = logaddexp(0, x) = max(x,0) + log1p(exp(-|x|))
__device__ __forceinline__ float softplus_f(float x) {
    return fmaxf(x, 0.0f) + log1pf(expf(-fabsf(x)));
}

// Full-wave32 sum via V_WMMA_F32_16X16X4_F32.
// A: per-lane {v, 0} -> each lane's value occupies a unique slot of the 16x4 A
// matrix, all other slots are 0. B = all ones (layout-irrelevant).
// D[m][n] = rowsum_m. Summing the 8 accumulator VGPRs gives a half-matrix sum
// per lane; one xor-16 shuffle combines halves -> total in every lane.
__device__ __forceinline__ float wave_reduce_wmma(float v) {
    v2f a; a.x = v; a.y = 0.0f;
    v2f b; b.x = 1.0f; b.y = 1.0f;
    v8f c = {};
    c = __builtin_amdgcn_wmma_f32_16x16x4_f32(false, a, false, b, (short)0, c,
                                              false, false);
    float s = c[0] + c[1] + c[2] + c[3] + c[4] + c[5] + c[6] + c[7];
    s += __shfl_xor(s, 16, 32);
    return s;
}

// Block reduction (blockDim.x multiple of 32, call with full EXEC).
// Result valid in thread 0. Fixed summation order -> deterministic.
__device__ __forceinline__ float block_reduce(float v, float* lds8) {
    float w = wave_reduce_wmma(v);
    int wave = threadIdx.x >> 5;
    int lane = threadIdx.x & 31;
    if (lane == 0) lds8[wave] = w;
    __syncthreads();
    float s = 0.0f;
    if (threadIdx.x == 0) {
        int nw = blockDim.x >> 5;
        for (int i = 0; i < nw; ++i) s += lds8[i];  // fixed order
    }
    __syncthreads();
    return s;
}

// ---- Kernel 0: reduce hann kernel -> S (sum) and Z (background zero count)
__global__ void hl_init_kernel(const float* __restrict__ hann, float* sZ) {
    __shared__ float sS[256];
    __shared__ int sN[256];
    float s = 0.0f; int nz = 0;
    for (int i = threadIdx.x; i < KK * KK; i += 256) {
        float h = hann[i];
        s += h;
        nz += (h != 0.0f) ? 1 : 0;
    }
    sS[threadIdx.x] = s; sN[threadIdx.x] = nz;
    __syncthreads();
    for (int off = 128; off > 0; off >>= 1) {
        if (threadIdx.x < off) {
            sS[threadIdx.x] += sS[threadIdx.x + off];
            sN[threadIdx.x] += sN[threadIdx.x + off];
        }
        __syncthreads();
    }
    if (threadIdx.x == 0) {
        sZ[0] = sS[0];
        sZ[1] = (float)(HH * WW - sN[0]);
    }
}

// ---- Kernel 1: locate the 33x33 ones-block corner with a sparse probe.
// Probes at stride 32 are guaranteed to hit a 33-wide block; then a 33-step
// parallel walk up/left finds the exact edges. ~300 loads/image vs 1M,
// avoiding the 48 MB full scan of target entirely.
__global__ void hl_corner_kernel(const float* __restrict__ tgt,
                                 int* __restrict__ ymin, int* __restrict__ xmin) {
    const int img = blockIdx.x;
    const float* t = tgt + (size_t)img * HH * WW;
    __shared__ int hitEnc, ry, rx;
    if (threadIdx.x == 0) { hitEnc = 0x7FFFFFFF; ry = 0x7FFFFFFF; rx = 0x7FFFFFFF; }
    __syncthreads();
    {
        int pr = (threadIdx.x >> 4) * 32;   // 0,32,...,480
        int pc = (threadIdx.x & 15) * 32;
        if (t[pr * WW + pc] == 1.0f) atomicMin(&hitEnc, (pr << 16) | pc);
    }
    __syncthreads();
    int r = hitEnc >> 16, c = hitEnc & 0xFFFF;
    if (threadIdx.x < KK) {
        int row = r - (int)threadIdx.x;
        if (row >= 0 && t[row * WW + c] == 1.0f) atomicMin(&ry, row);
    } else if (threadIdx.x >= 64 && threadIdx.x < 64 + KK) {
        int col = c - (int)(threadIdx.x - 64);
        if (col >= 0 && t[r * WW + col] == 1.0f) atomicMin(&rx, col);
    }
    __syncthreads();
    if (threadIdx.x == 0) { ymin[img] = ry; xmin[img] = rx; }
}

// ---- Kernel 2: streaming pass over pred only: sum softplus(pred).
// 48 MB of global_load_b128, bandwidth bound (~2.1 us at 23.3 TB/s). Every
// thread runs exactly n4/stride iterations (n4 divisible by grid*block), so
// EXEC is full at the WMMA reduce.
__global__ void hl_pass1_kernel(const float4* __restrict__ pred4,
                                float* __restrict__ partials, int n4) {
    float acc = 0.0f;
    const int stride = gridDim.x * blockDim.x;
    for (int i = blockIdx.x * blockDim.x + threadIdx.x; i < n4; i += stride) {
        __builtin_prefetch(pred4 + i + stride, 0, 0);  // global_prefetch_b8, speculative -> OOB-safe
        float4 p = pred4[i];
        acc += softplus_f(p.x) + softplus_f(p.y) + softplus_f(p.z) + softplus_f(p.w);
    }
    __shared__ float lds8[8];
    float bs = block_reduce(acc, lds8);
    if (threadIdx.x == 0) partials[blockIdx.x] = bs;
}

// ---- Kernel 3: window correction per image.
// Pass1 charged every pixel weight 1/(2Z) on softplus only. Inside the window
// (where target==1): hann!=0 pixels actually weigh hk/(2S) with bce=sp-p;
// hann==0 pixels keep 1/(2Z) but are missing the -p term.
__global__ void hl_pass2_kernel(const float* __restrict__ pred,
                                const float* __restrict__ hann,
                                const int* __restrict__ ymin,
                                const int* __restrict__ xmin,
                                const float* __restrict__ sZ,
                                float* __restrict__ Cws) {
    const int img = blockIdx.x;
    const int y0 = ymin[img], x0 = xmin[img];
    const float inv2S = 0.5f / sZ[0];
    const float inv2Z = 0.5f / sZ[1];
    const float* p = pred + (size_t)img * HH * WW;
    float acc = 0.0f;
    for (int e = threadIdx.x; e < KK * KK; e += blockDim.x) {
        int dy = e / KK;
        int dx = e - dy * KK;
        float pv = p[(y0 + dy) * WW + (x0 + dx)];
        float hk = hann[e];
        float sp = softplus_f(pv);
        acc += (hk != 0.0f) ? ((sp - pv) * hk * inv2S - sp * inv2Z)
                            : (-pv * inv2Z);
    }
    __shared__ float lds8[8];
    float bs = block_reduce(acc, lds8);
    if (threadIdx.x == 0) Cws[img] = bs;
}

// ---- Kernel 4: deterministic finalize in double.
__global__ void hl_final_kernel(const float* __restrict__ partials, int nP,
                                const float* __restrict__ Cws,
                                const float* __restrict__ sZ,
                                float* __restrict__ out) {
    __shared__ double dlds[256];
    double s = 0.0;
    for (int i = threadIdx.x; i < nP; i += 256) s += (double)partials[i];
    dlds[threadIdx.x] = s;
    __syncthreads();
    for (int off = 128; off > 0; off >>= 1) {
        if (threadIdx.x < off) dlds[threadIdx.x] += dlds[threadIdx.x + off];
        __syncthreads();
    }
    if (threadIdx.x == 0) {
        double sc = 0.0;
        for (int i = 0; i < BB; ++i) sc += (double)Cws[i];
        double inv2Z = 0.5 / (double)sZ[1];
        out[0] = (float)((dlds[0] * inv2Z + sc) / (double)BB);
    }
}

extern "C" void kernel_launch(void* const* d_in, const int* in_sizes, int n_in,
                              void* d_out, int out_size, void* d_ws, size_t ws_size,
                              hipStream_t stream) {
    const float* pred   = (const float*)d_in[0];   // [48,512,512] f32
    const float* target = (const float*)d_in[1];   // [48,512,512] f32
    const float* hann   = (const float*)d_in[2];   // [33,33] f32
    float* out = (float*)d_out;

    // workspace layout (floats/ints, ~9 KB)
    float* sZ       = (float*)d_ws;              // [0]=S, [1]=Z
    int*   ymin     = (int*)d_ws + 2;            // 48
    int*   xmin     = (int*)d_ws + 2 + BB;       // 48
    float* partials = (float*)d_ws + 128;        // 2048
    float* Cws      = (float*)d_ws + 128 + 2048; // 48

    const int n4 = in_sizes[0] / 4;              // 3,145,728 float4 elements
    const int NBLK = 2048;                       // n4 % (2048*256) == 0

    hl_init_kernel  <<<1,    256, 0, stream>>>(hann, sZ);
    hl_corner_kernel<<<BB,   256, 0, stream>>>(target, ymin, xmin);
    hl_pass1_kernel <<<NBLK, 256, 0, stream>>>((const float4*)pred, partials, n4);
    hl_pass2_kernel <<<BB,   256, 0, stream>>>(pred, hann, ymin, xmin, sZ, Cws);
    hl_final_kernel <<<1,    256, 0, stream>>>(partials, NBLK, Cws, sZ, out);
}